// TSP_GNN_45054206935572
// MI455X (gfx1250) — compile-verified
//
#include <hip/hip_runtime.h>
#include <hip/hip_bf16.h>

// ---------------------------------------------------------------------------
// GCN forward for MI455X (gfx1250, wave32).
//  - All GEMMs: V_WMMA_F32_16X16X4_F32 (exact fp32 matrix op)
//  - Tile staging: double-buffered GLOBAL_LOAD_ASYNC_TO_LDS_B128 (ASYNCcnt)
// ---------------------------------------------------------------------------

typedef __attribute__((ext_vector_type(2))) float v2f;
typedef __attribute__((ext_vector_type(8))) float v8f;

#define NN 8192      // nodes (feature width of layer-0 input and fc output)
#define NE 262144    // edges
#define HD 64        // hidden dim

// LDS tile geometry (row pitch 68 floats = 272 B = 17*16 B -> b128-aligned,
// +4 float pad keeps column-strided reads bank-conflict free)
#define A_ROWS 32
#define B_ROWS 64
#define PITCH  68
#define A_TILE_BYTES (A_ROWS * PITCH * 4)             // 8704
#define TILE_BYTES   ((A_ROWS + B_ROWS) * PITCH * 4)  // 26112
#define SMEM_BYTES   (2 * TILE_BYTES)                 // 52224 (double buffer)

// Async global->LDS copy of 16 bytes (per-lane). GV addressing mode.
__device__ __forceinline__ void async_b128_to_lds(unsigned lds_addr, const void* gptr) {
    asm volatile("global_load_async_to_lds_b128 %0, %1, off"
                 :: "v"(lds_addr), "v"(gptr) : "memory");
}
__device__ __forceinline__ void wait_async0() {
    asm volatile("s_wait_asynccnt 0x0" ::: "memory");
}

// ---------------------------------------------------------------------------
// Degree with self loops: deg = 1.0 (self loop) + sum of edge weights at dst
// ---------------------------------------------------------------------------
__global__ void deg_init_kernel(float* __restrict__ deg) {
    int i = blockIdx.x * blockDim.x + threadIdx.x;
    if (i < NN) deg[i] = 1.0f;
}

__global__ void deg_acc_kernel(const int* __restrict__ dst,
                               const float* __restrict__ ew,
                               float* __restrict__ deg) {
    int e = blockIdx.x * blockDim.x + threadIdx.x;
    if (e < NE) unsafeAtomicAdd(deg + dst[e], ew[e]);
}

__global__ void deg_rsqrt_kernel(float* __restrict__ deg) {
    int i = blockIdx.x * blockDim.x + threadIdx.x;
    if (i < NN) deg[i] = rsqrtf(deg[i]);
}

// ---------------------------------------------------------------------------
// WMMA fp32 GEMM:  C[M x (64*gridDim.y)] = A[M x K] @ B[K x ldb] (+bias)
//   256 threads = 8 wave32s (2x4), each wave owns a 16x16 f32 accumulator.
//   Block tile 32x64, K staged 64 at a time through double-buffered LDS
//   filled with GLOBAL_LOAD_ASYNC_TO_LDS_B128; next tile's loads are in
//   flight while the current tile's 16 WMMAs execute.
// EXEC is all-ones throughout (required by WMMA).
// ---------------------------------------------------------------------------
__global__ __launch_bounds__(256)
void gemm_wmma_f32_kernel(const float* __restrict__ A,
                          const float* __restrict__ B,
                          float* __restrict__ C,
                          const float* __restrict__ bias, // may be null
                          int M, int K, int ldb, int ldc) {
    extern __shared__ char smem[];
    // generic LDS pointer low 32 bits == LDS byte offset (ISA 10.2 aperture calc)
    const unsigned lds_base = (unsigned)(size_t)smem;

    const int tid  = threadIdx.x;
    const int lane = tid & 31;
    const int wave = tid >> 5;      // 0..7
    const int wm   = wave >> 2;     // 0..1  (row block of wave)
    const int wn   = wave & 3;      // 0..3  (col block of wave)

    const int m0  = blockIdx.x * 32;       // block row origin
    const int nt0 = blockIdx.y * 64;       // block col origin

    const int l16   = lane & 15;
    const int lhalf = lane >> 4;           // 0: lanes 0-15, 1: lanes 16-31
    const int row_a = wm * 16 + l16;       // A fragment row (M)
    const int kh    = lhalf * 2;           // A/B fragment K sub-offset (ISA layout)
    const int col_b = wn * 16 + l16;       // B/C fragment col (N)

    // issue one K-tile (A: 32x64, B: 64x64) as 6 async b128 loads per thread
    auto issue_tile = [&](int k0, int bufsel) {
        const unsigned abase = lds_base + bufsel * TILE_BYTES;
        const unsigned bbase = abase + A_TILE_BYTES;
        #pragma unroll
        for (int i = tid; i < A_ROWS * 16; i += 256) {      // 512 b128 chunks
            int r  = i >> 4;
            int c4 = (i & 15) << 2;
            async_b128_to_lds(abase + (unsigned)(r * PITCH + c4) * 4,
                              A + (size_t)(m0 + r) * K + k0 + c4);
        }
        #pragma unroll
        for (int i = tid; i < B_ROWS * 16; i += 256) {      // 1024 b128 chunks
            int r  = i >> 4;
            int c4 = (i & 15) << 2;
            async_b128_to_lds(bbase + (unsigned)(r * PITCH + c4) * 4,
                              B + (size_t)(k0 + r) * ldb + nt0 + c4);
        }
    };

    v8f acc = {};

    issue_tile(0, 0);
    wait_async0();
    __syncthreads();

    int buf = 0;
    for (int k0 = 0; k0 < K; k0 += 64) {
        if (k0 + 64 < K) issue_tile(k0 + 64, buf ^ 1);  // overlap with compute

        const float* As = (const float*)(smem + buf * TILE_BYTES);
        const float* Bs = As + A_ROWS * PITCH;

        #pragma unroll
        for (int kk = 0; kk < 64; kk += 4) {
            // A 16x4 fp32 fragment: lanes 0-15 hold K=kk,kk+1; lanes 16-31 kk+2,kk+3
            v2f a, b;
            a.x = As[row_a * PITCH + kk + kh];
            a.y = As[row_a * PITCH + kk + kh + 1];
            // B 4x16 fp32 fragment: row striped across lanes within a VGPR
            b.x = Bs[(kk + kh) * PITCH + col_b];
            b.y = Bs[(kk + kh + 1) * PITCH + col_b];
            acc = __builtin_amdgcn_wmma_f32_16x16x4_f32(
                /*neg_a=*/false, a, /*neg_b=*/false, b,
                /*c_mod=*/(short)0, acc, /*reuse_a=*/false, /*reuse_b=*/false);
        }

        wait_async0();       // drain this wave's async loads (next tile landed)
        __syncthreads();     // all waves done reading current buffer
        buf ^= 1;
    }

    // C/D 16x16 f32 layout: VGPR r -> M=r (lanes 0-15), M=r+8 (lanes 16-31)
    const int ncol = nt0 + col_b;
    const int mb   = wm * 16 + lhalf * 8;
    float badd = bias ? bias[ncol] : 0.0f;
    #pragma unroll
    for (int r = 0; r < 8; ++r) {
        C[(size_t)(m0 + mb + r) * ldc + ncol] = acc[r] + badd;
    }
}

// ---------------------------------------------------------------------------
// Edge scatter:  out[dst] += x[src] * (dinv[src]*w*dinv[dst])
// Self loops handled implicitly as edges [NE, NE+NN) with weight 1.
// One thread per (edge, column). out must be pre-zeroed.
// ---------------------------------------------------------------------------
__global__ void scatter_kernel(const int* __restrict__ src,
                               const int* __restrict__ dst,
                               const float* __restrict__ ew,
                               const float* __restrict__ dinv,
                               const float* __restrict__ x,
                               float* __restrict__ out) {
    unsigned gid = blockIdx.x * blockDim.x + threadIdx.x;
    int e   = (int)(gid >> 6);
    int col = (int)(gid & 63);
    if (e >= NE + NN) return;
    int s, d; float w;
    if (e < NE) { s = src[e]; d = dst[e]; w = ew[e]; }
    else        { s = d = e - NE;         w = 1.0f;  }
    float norm = dinv[s] * w * dinv[d];
    float v = x[(size_t)s * HD + col] * norm;
    unsafeAtomicAdd(out + (size_t)d * HD + col, v);
}

__global__ void zero_kernel(float* __restrict__ p, int n) {
    int i = blockIdx.x * blockDim.x + threadIdx.x;
    if (i < n) p[i] = 0.0f;
}

__global__ void relu_bias_kernel(float* __restrict__ x,
                                 const float* __restrict__ b, int n) {
    int i = blockIdx.x * blockDim.x + threadIdx.x;
    if (i < n) {
        float v = x[i] + b[i & (HD - 1)];
        x[i] = v > 0.0f ? v : 0.0f;
    }
}

// ---------------------------------------------------------------------------
// In-place row softmax; one 256-thread block per 8192-wide row
// ---------------------------------------------------------------------------
__global__ __launch_bounds__(256)
void softmax_rows_kernel(float* __restrict__ out, int ncols) {
    __shared__ float red[256];
    float* row = out + (size_t)blockIdx.x * ncols;
    const int tid = threadIdx.x;

    float m = -3.402823466e38f;
    for (int c = tid; c < ncols; c += 256) m = fmaxf(m, row[c]);
    red[tid] = m; __syncthreads();
    for (int s = 128; s > 0; s >>= 1) {
        if (tid < s) red[tid] = fmaxf(red[tid], red[tid + s]);
        __syncthreads();
    }
    m = red[0]; __syncthreads();

    float sum = 0.0f;
    for (int c = tid; c < ncols; c += 256) sum += __expf(row[c] - m);
    red[tid] = sum; __syncthreads();
    for (int s = 128; s > 0; s >>= 1) {
        if (tid < s) red[tid] += red[tid + s];
        __syncthreads();
    }
    float inv = 1.0f / red[0];

    for (int c = tid; c < ncols; c += 256) row[c] = __expf(row[c] - m) * inv;
}

// ---------------------------------------------------------------------------
// Driver
// ---------------------------------------------------------------------------
extern "C" void kernel_launch(void* const* d_in, const int* in_sizes, int n_in,
                              void* d_out, int out_size, void* d_ws, size_t ws_size,
                              hipStream_t stream) {
    const int*   edge_index = (const int*)d_in[0];     // [2, E] flat
    const int*   src  = edge_index;
    const int*   dst  = edge_index + NE;
    const float* ew   = (const float*)d_in[1];         // [E]
    const float* X    = (const float*)d_in[2];         // [N, N]
    const float* W1   = (const float*)d_in[3];         // [N, H]
    const float* b1   = (const float*)d_in[4];         // [H]
    const float* W2   = (const float*)d_in[5];         // [H, H]
    const float* b2   = (const float*)d_in[6];         // [H]
    const float* fcW  = (const float*)d_in[7];         // [H, N]
    const float* fcb  = (const float*)d_in[8];         // [N]
    float*       out  = (float*)d_out;                 // [N, N]

    // workspace: dinv (N) | bufA (N*H) | bufB (N*H)  ~= 4.2 MB
    float* dinv = (float*)d_ws;
    float* bufA = dinv + NN;
    float* bufB = bufA + (size_t)NN * HD;

    const int T = 256;
    const int nh = NN * HD;

    // symmetric-normalization coefficients
    deg_init_kernel <<<NN / T, T, 0, stream>>>(dinv);
    deg_acc_kernel  <<<NE / T, T, 0, stream>>>(dst, ew, dinv);
    deg_rsqrt_kernel<<<NN / T, T, 0, stream>>>(dinv);

    // layer 1: x1 = X @ W1  (K = 8192)
    gemm_wmma_f32_kernel<<<dim3(NN / 32, 1), T, SMEM_BYTES, stream>>>(
        X, W1, bufA, nullptr, NN, NN, HD, HD);
    zero_kernel<<<nh / T, T, 0, stream>>>(bufB, nh);
    scatter_kernel<<<((NE + NN) * HD + T - 1) / T, T, 0, stream>>>(
        src, dst, ew, dinv, bufA, bufB);
    relu_bias_kernel<<<nh / T, T, 0, stream>>>(bufB, b1, nh);

    // layer 2: x2 = h1 @ W2  (K = 64)
    gemm_wmma_f32_kernel<<<dim3(NN / 32, 1), T, SMEM_BYTES, stream>>>(
        bufB, W2, bufA, nullptr, NN, HD, HD, HD);
    zero_kernel<<<nh / T, T, 0, stream>>>(bufB, nh);
    scatter_kernel<<<((NE + NN) * HD + T - 1) / T, T, 0, stream>>>(
        src, dst, ew, dinv, bufA, bufB);
    relu_bias_kernel<<<nh / T, T, 0, stream>>>(bufB, b2, nh);

    // fc: out = h2 @ fcW + fcb  (K = 64, wide N)
    gemm_wmma_f32_kernel<<<dim3(NN / 32, NN / 64), T, SMEM_BYTES, stream>>>(
        bufB, fcW, out, fcb, NN, HD, NN, NN);

    // row softmax in place
    softmax_rows_kernel<<<NN, T, 0, stream>>>(out, NN);
}